// GBSWT_42382737277303
// MI455X (gfx1250) — compile-verified
//
#include <hip/hip_runtime.h>
#include <hip/hip_bf16.h>
#include <math.h>
#include <float.h>

#define BB 8
#define SS 2048
#define DIMD 768
#define KW 9
#define SP 2052
#define SUPER 36
#define NSB (SP / SUPER) /* 57 */
#define MT 32            /* GEMM rows per block */

typedef float v2f __attribute__((ext_vector_type(2)));
typedef float v8f __attribute__((ext_vector_type(8)));

// ---------------- K0: pack pj_w (o,i) -> k-pair-interleaved Wt:
// wt2 float-pairs: pair p=i>>1, col o:  wt2[(p*DIMD+o)*2 + (i&1)] = W[o][i]
// so a B-fragment {B[2p][n], B[2p+1][n]} is ONE aligned 8-byte load.
__global__ void k_pack(const float* __restrict__ w, float* __restrict__ wt2) {
  int idx = blockIdx.x * blockDim.x + threadIdx.x;
  if (idx < DIMD * DIMD) {
    int o = idx / DIMD, i = idx % DIMD;
    wt2[((i >> 1) * DIMD + o) * 2 + (i & 1)] = w[idx];
  }
}

// ---------------- K1: embedding gather + depthwise conv (k=9) + bias
__global__ void k_embed_dw(const int* __restrict__ tokens,
                           const float* __restrict__ emb,
                           const float* __restrict__ dww,
                           const float* __restrict__ dwb,
                           float* __restrict__ y) {
  int bt = blockIdx.x;              // 0 .. B*S-1
  int b = bt / SS, t = bt % SS;
  __shared__ int tok[KW];
  if (threadIdx.x < KW) {
    int tt = t + threadIdx.x;
    tok[threadIdx.x] = (tt < SS) ? tokens[b * SS + tt] : -1;
  }
  __syncthreads();
  for (int c = threadIdx.x; c < DIMD; c += blockDim.x) {
    float acc = dwb[c];
#pragma unroll
    for (int k = 0; k < KW; ++k) {
      int tk = tok[k];
      if (tk >= 0) acc += emb[tk * DIMD + c] * dww[c * KW + k];
    }
    y[(long long)bt * DIMD + c] = acc;
  }
}

// ---------------- K2: in-place GEMM  z[r,:] = y[r,:] @ W^T + bias  (f32 WMMA)
// 32 rows per block staged in LDS via async-to-LDS; each wave owns a 32x32
// output tile (both row groups) so each B-fragment feeds 2 WMMAs.
__global__ void __launch_bounds__(256) k_gemm(float* __restrict__ z,
                                              const float* __restrict__ wt2,
                                              const float* __restrict__ bias) {
  __shared__ float a_lds[MT][DIMD + 4];
  int row0 = blockIdx.x * MT;

  // async stage: 32x768 f32 tile, 16B per lane per issue (ASYNCcnt-tracked)
  for (int idx = threadIdx.x * 4; idx < MT * DIMD; idx += 256 * 4) {
    int r = idx / DIMD, c = idx % DIMD;   // c is a multiple of 4
    unsigned lv = (unsigned)(unsigned long long)&a_lds[r][c];
    const float* g = z + (long long)(row0 + r) * DIMD + c;
    asm volatile("global_load_async_to_lds_b128 %0, %1, off" :: "v"(lv), "v"(g) : "memory");
  }
  asm volatile("s_wait_asynccnt 0x0" ::: "memory");
  __syncthreads();

  int lane = threadIdx.x & 31;
  int wave = threadIdx.x >> 5;     // 0..7 -> 32-col strip
  int h = lane >> 4;               // K half selector
  int l15 = lane & 15;

  for (int nc = 0; nc < 3; ++nc) {
    int colbase = nc * 256 + wave * 32;
    float b0 = bias[colbase + l15];
    float b1 = bias[colbase + 16 + l15];
    v8f acc00, acc01, acc10, acc11;
#pragma unroll
    for (int r = 0; r < 8; ++r) { acc00[r] = b0; acc01[r] = b1; acc10[r] = b0; acc11[r] = b1; }

    for (int k0 = 0; k0 < DIMD; k0 += 4) {
      v2f a0, a1;
      a0.x = a_lds[l15][k0 + 2 * h];
      a0.y = a_lds[l15][k0 + 2 * h + 1];
      a1.x = a_lds[16 + l15][k0 + 2 * h];
      a1.y = a_lds[16 + l15][k0 + 2 * h + 1];
      int p = (k0 >> 1) + h;       // k-pair index
      v2f bf0 = *(const v2f*)(wt2 + ((long long)p * DIMD + colbase + l15) * 2);
      v2f bf1 = *(const v2f*)(wt2 + ((long long)p * DIMD + colbase + 16 + l15) * 2);
      acc00 = __builtin_amdgcn_wmma_f32_16x16x4_f32(false, a0, false, bf0, (short)0, acc00, false, false);
      acc01 = __builtin_amdgcn_wmma_f32_16x16x4_f32(false, a0, false, bf1, (short)0, acc01, false, false);
      acc10 = __builtin_amdgcn_wmma_f32_16x16x4_f32(false, a1, false, bf0, (short)0, acc10, false, false);
      acc11 = __builtin_amdgcn_wmma_f32_16x16x4_f32(false, a1, false, bf1, (short)0, acc11, false, false);
    }
#pragma unroll
    for (int r = 0; r < 8; ++r) {
      int g0 = row0 + r + 8 * h;
      int g1 = row0 + 16 + r + 8 * h;
      z[(long long)g0 * DIMD + colbase + l15] = acc00[r];
      z[(long long)g0 * DIMD + colbase + 16 + l15] = acc01[r];
      z[(long long)g1 * DIMD + colbase + l15] = acc10[r];
      z[(long long)g1 * DIMD + colbase + 16 + l15] = acc11[r];
    }
  }
}

// ---------------- K3: u[b,p] = z[b,p,:] . score_w   (padded rows -> 0)
__global__ void k_u(const float* __restrict__ z, const float* __restrict__ sw,
                    float* __restrict__ u) {
  int rid = blockIdx.x * 8 + (threadIdx.x >> 5);
  int lane = threadIdx.x & 31;
  if (rid >= BB * SP) return;
  int b = rid / SP, p = rid % SP;
  float acc = 0.f;
  if (p < SS) {
    const float* zr = z + (long long)(b * SS + p) * DIMD;
    for (int c = lane; c < DIMD; c += 32) acc += zr[c] * sw[c];
  }
  for (int off = 16; off; off >>= 1) acc += __shfl_down(acc, off, 32);
  if (lane == 0) u[rid] = acc;
}

// ---------------- K4: per-position 6-scale block scores + softmax over scales
__global__ void k_scores(const float* __restrict__ u, const unsigned char* __restrict__ mask,
                         const float* __restrict__ sbptr, float* __restrict__ s) {
  int g = blockIdx.x % NSB;
  int b = blockIdx.x / NSB;
  __shared__ float ul[SUPER];
  __shared__ float ml[SUPER];
  int t = threadIdx.x;
  if (t < SUPER) {
    int p = g * SUPER + t;
    ul[t] = u[b * SP + p];
    ml[t] = (p < SS && mask[b * SS + p]) ? 1.f : 0.f;
  }
  __syncthreads();
  if (t < SUPER) {
    const int bsl[6] = {1, 2, 3, 4, 6, 9};
    float sb = sbptr[0];
    float pre[6];
#pragma unroll
    for (int n = 0; n < 6; ++n) {
      int bs = bsl[n];
      int st = (t / bs) * bs;
      float su = 0.f, cnt = 0.f;
      for (int j = 0; j < bs; ++j) { su += ul[st + j] * ml[st + j]; cnt += ml[st + j]; }
      pre[n] = (cnt > 0.f) ? (su / fmaxf(cnt, 1.f) + sb) : -FLT_MAX;
    }
    float mx = pre[0];
#pragma unroll
    for (int n = 1; n < 6; ++n) mx = fmaxf(mx, pre[n]);
    float se = 0.f, e[6];
#pragma unroll
    for (int n = 0; n < 6; ++n) { e[n] = expf(pre[n] - mx); se += e[n]; }
    int p = g * SUPER + t;
#pragma unroll
    for (int n = 0; n < 6; ++n) s[(long long)(b * SP + p) * 6 + n] = e[n] / se;
  }
}

// ---------------- K6: score attention, one wave32 per query row (no barriers
// in the hot loop; wave shuffles for the max / sum reductions)
__global__ void __launch_bounds__(256) k_attn(const float* __restrict__ s,
                                              const unsigned char* __restrict__ mask,
                                              float* __restrict__ os) {
  int g = blockIdx.x % NSB;
  int b = blockIdx.x / NSB;
  __shared__ float sl[SP * 6];
  __shared__ unsigned char ml[SP];
  int t = threadIdx.x;
  for (int idx = t; idx < SP * 6; idx += 256) sl[idx] = s[(long long)b * SP * 6 + idx];
  for (int j = t; j < SP; j += 256) ml[j] = (j < SS) ? mask[b * SS + j] : 0;
  __syncthreads();

  int wave = t >> 5, lane = t & 31;
  for (int it = 0; it < 5; ++it) {
    int i = it * 8 + wave;
    if (i >= SUPER) continue;
    int p = g * SUPER + i;
    if (p >= SS) continue;        // outputs beyond S never used
    if (!ml[p]) continue;         // masked query: final output zeroed anyway
    float mi[6];
#pragma unroll
    for (int n = 0; n < 6; ++n) mi[n] = sl[p * 6 + n];
    // pass 1: row max
    float mx = -FLT_MAX;
    for (int j = lane; j < SP; j += 32) {
      if (ml[j]) {
        float sim = 0.f;
#pragma unroll
        for (int n = 0; n < 6; ++n) sim += mi[n] * sl[j * 6 + n];
        mx = fmaxf(mx, sim);
      }
    }
#pragma unroll
    for (int off = 16; off; off >>= 1) mx = fmaxf(mx, __shfl_xor(mx, off, 32));
    // pass 2: sum exp and weighted sum of s
    float se = 0.f, sv[6] = {0, 0, 0, 0, 0, 0};
    for (int j = lane; j < SP; j += 32) {
      if (ml[j]) {
        float sim = 0.f;
#pragma unroll
        for (int n = 0; n < 6; ++n) sim += mi[n] * sl[j * 6 + n];
        float e = expf(sim - mx);
        se += e;
#pragma unroll
        for (int n = 0; n < 6; ++n) sv[n] += e * sl[j * 6 + n];
      }
    }
#pragma unroll
    for (int off = 16; off; off >>= 1) {
      se += __shfl_xor(se, off, 32);
#pragma unroll
      for (int n = 0; n < 6; ++n) sv[n] += __shfl_xor(sv[n], off, 32);
    }
    float outv = (lane == 0) ? sv[0] : (lane == 1) ? sv[1] : (lane == 2) ? sv[2]
               : (lane == 3) ? sv[3] : (lane == 4) ? sv[4] : sv[5];
    if (lane < 6) os[(long long)(b * SP + p) * 6 + lane] = outv / se;
  }
}

// ---------------- K7: out[i,:] = mask_i * sum_n w[i][n] * blockmean_n(i)
__global__ void __launch_bounds__(256) k_out(const float* __restrict__ z,
                                             const float* __restrict__ os,
                                             const unsigned char* __restrict__ mask,
                                             float* __restrict__ out,
                                             float* __restrict__ omask) {
  int g = blockIdx.x % NSB;
  int b = blockIdx.x / NSB;
  __shared__ float wm[SUPER][6];
  __shared__ float ml[SUPER];
  int t = threadIdx.x;
  if (t < SUPER) {
    int p = g * SUPER + t;
    ml[t] = (p < SS && mask[b * SS + p]) ? 1.f : 0.f;
  }
  __syncthreads();
  if (t < SUPER * 6) {
    int i = t / 6, n = t % 6;
    const int bsl[6] = {1, 2, 3, 4, 6, 9};
    int bs = bsl[n];
    int st = (i / bs) * bs;
    float cnt = 0.f;
    for (int j = 0; j < bs; ++j) cnt += ml[st + j];
    int p = g * SUPER + i;
    wm[i][n] = os[(long long)(b * SP + p) * 6 + n] / fmaxf(cnt, 1.f);
  }
  __syncthreads();
  if (t < SUPER) {
    int p = g * SUPER + t;
    if (p < SS) omask[b * SS + p] = ml[t];
  }
  for (int d = t; d < DIMD; d += 256) {
    float zr[SUPER];
#pragma unroll
    for (int j = 0; j < SUPER; ++j) {
      int p = g * SUPER + j;
      zr[j] = (p < SS && ml[j] > 0.f) ? z[(long long)(b * SS + p) * DIMD + d] : 0.f;
    }
    float s2[18], s3[12], s4[9], s6[6], s9[4];
#pragma unroll
    for (int k = 0; k < 18; ++k) s2[k] = zr[2 * k] + zr[2 * k + 1];
#pragma unroll
    for (int k = 0; k < 12; ++k) s3[k] = zr[3 * k] + zr[3 * k + 1] + zr[3 * k + 2];
#pragma unroll
    for (int k = 0; k < 9; ++k) s4[k] = s2[2 * k] + s2[2 * k + 1];
#pragma unroll
    for (int k = 0; k < 6; ++k) s6[k] = s3[2 * k] + s3[2 * k + 1];
#pragma unroll
    for (int k = 0; k < 4; ++k) s9[k] = s3[3 * k] + s3[3 * k + 1] + s3[3 * k + 2];
#pragma unroll
    for (int i = 0; i < SUPER; ++i) {
      int p = g * SUPER + i;
      if (p < SS) {
        float val = wm[i][0] * zr[i] + wm[i][1] * s2[i / 2] + wm[i][2] * s3[i / 3] +
                    wm[i][3] * s4[i / 4] + wm[i][4] * s6[i / 6] + wm[i][5] * s9[i / 9];
        out[(long long)(b * SS + p) * DIMD + d] = val * ml[i];
      }
    }
  }
}

extern "C" void kernel_launch(void* const* d_in, const int* in_sizes, int n_in,
                              void* d_out, int out_size, void* d_ws, size_t ws_size,
                              hipStream_t stream) {
  const int* tokens = (const int*)d_in[0];
  const unsigned char* mask = (const unsigned char*)d_in[1];
  const float* emb = (const float*)d_in[2];
  const float* dww = (const float*)d_in[3];
  const float* dwb = (const float*)d_in[4];
  const float* pjw = (const float*)d_in[5];
  const float* pjb = (const float*)d_in[6];
  const float* sw = (const float*)d_in[7];
  const float* sb = (const float*)d_in[8];

  float* ws = (float*)d_ws;
  float* zbuf = ws;                                  // B*S*D   (y then z, in place)
  float* wt2 = zbuf + (size_t)BB * SS * DIMD;        // D*D (k-pair-interleaved W^T)
  float* ubuf = wt2 + (size_t)DIMD * DIMD;           // B*Sp
  float* sbuf = ubuf + (size_t)BB * SP;              // B*Sp*6
  float* osbuf = sbuf + (size_t)BB * SP * 6;         // B*Sp*6

  float* out = (float*)d_out;
  float* omask = out + (size_t)BB * SS * DIMD;

  k_pack<<<(DIMD * DIMD + 255) / 256, 256, 0, stream>>>(pjw, wt2);
  k_embed_dw<<<BB * SS, 256, 0, stream>>>(tokens, emb, dww, dwb, zbuf);
  k_gemm<<<BB * SS / MT, 256, 0, stream>>>(zbuf, wt2, pjb);
  k_u<<<(BB * SP) / 8, 256, 0, stream>>>(zbuf, sw, ubuf);
  k_scores<<<BB * NSB, 64, 0, stream>>>(ubuf, mask, sb, sbuf);
  k_attn<<<BB * NSB, 256, 0, stream>>>(sbuf, mask, osbuf);
  k_out<<<BB * NSB, 256, 0, stream>>>(zbuf, osbuf, mask, out, omask);
}